// Gram_18614388261662
// MI455X (gfx1250) — compile-verified
//
#include <hip/hip_runtime.h>
#include <hip/hip_bf16.h>
#include <stdint.h>

// ---------- types ----------
typedef __attribute__((ext_vector_type(2))) float v2f;
typedef __attribute__((ext_vector_type(4))) float v4f;
typedef __attribute__((ext_vector_type(8))) float v8f;
typedef __attribute__((ext_vector_type(4))) int   v4i;

typedef __attribute__((address_space(1))) v4i* gv4ip;   // global int4*
typedef __attribute__((address_space(3))) v4i* lv4ip;   // LDS int4*

// ---------- config ----------
#define C_DIM    512
#define BM       128      // output tile is BM x BM
#define KC       16       // K chunk per LDS stage
#define LDA      20       // KC + 4 pad: conflict-free ds_load_b64 across 16 rows, 16B row align
#define NTHREADS 256      // 8 wave32

#if defined(__has_builtin)
#if __has_builtin(__builtin_amdgcn_global_load_async_to_lds_b128)
#define HAVE_ASYNC_LDS 1
#endif
#endif

__device__ __forceinline__ void copy_b128_to_lds(const float* g, float* l) {
#ifdef HAVE_ASYNC_LDS
    __builtin_amdgcn_global_load_async_to_lds_b128((gv4ip)g, (lv4ip)l, 0, 0);
#else
    *(v4f*)l = *(const v4f*)g;
#endif
}

__device__ __forceinline__ void wait_async_all() {
#ifdef HAVE_ASYNC_LDS
#if __has_builtin(__builtin_amdgcn_s_wait_asynccnt)
    __builtin_amdgcn_s_wait_asynccnt(0);
#else
    asm volatile("s_wait_asynccnt 0x0" ::: "memory");
#endif
#endif
}

// ---------- main gram kernel ----------
// Each block computes a 128x128 tile of G over K range [blockIdx.y*kPerSplit, +kPerSplit),
// writing a partial (or final, if split==1) tile to out + blockIdx.y*C*C.
__global__ __launch_bounds__(NTHREADS)
void gram_wmma_kernel(const float* __restrict__ X, float* __restrict__ out,
                      int npix, int kPerSplit) {
    __shared__ float lsA[2][BM * LDA];
    __shared__ float lsB[2][BM * LDA];

    const int bt = blockIdx.x;            // 0..15
    const int ti = bt >> 2;               // row tile
    const int tj = bt & 3;                // col tile
    const int kbase   = blockIdx.y * kPerSplit;
    const int nChunks = kPerSplit / KC;

    const int t    = threadIdx.x;
    const int lane = t & 31;
    const int wave = t >> 5;              // 0..7
    const int wm   = wave >> 2;           // 0..1 : 64-row slab
    const int wn   = wave & 3;            // 0..3 : 32-col slab
    const int half = lane >> 4;           // 0/1
    const int r    = lane & 15;

    const float* Xa = X + (size_t)(ti * BM) * npix;
    const float* Xb = X + (size_t)(tj * BM) * npix;

    // cooperative loader: 128 rows x 4 float4-cols per tile = 512 granules; 2 per thread
    auto loadTile = [&](int kc, int buf) {
        const int k0 = kbase + kc * KC;
#pragma unroll
        for (int p = 0; p < 2; ++p) {
            const int g   = p * NTHREADS + t;
            const int row = g >> 2;
            const int c4  = (g & 3) << 2;
            copy_b128_to_lds(Xa + (size_t)row * npix + k0 + c4, &lsA[buf][row * LDA + c4]);
            copy_b128_to_lds(Xb + (size_t)row * npix + k0 + c4, &lsB[buf][row * LDA + c4]);
        }
    };

    v8f acc[4][2];
#pragma unroll
    for (int m = 0; m < 4; ++m)
#pragma unroll
        for (int n = 0; n < 2; ++n)
            acc[m][n] = {};

    loadTile(0, 0);

    for (int kc = 0; kc < nChunks; ++kc) {
        const int cur = kc & 1;
        wait_async_all();         // this wave's fills of buffer `cur` are in LDS
        __syncthreads();          // everyone's fills done + everyone done computing buf cur^1

        if (kc + 1 < nChunks) loadTile(kc + 1, cur ^ 1);

        // L2 prefetch ~8 chunks (512B) ahead: one cacheline per thread covers both tiles
        {
            const int kp = kbase + (kc + 8) * KC;
            if (kp < kbase + kPerSplit) {
                const float* pf = ((t & 1) ? Xb : Xa) + (size_t)(t >> 1) * npix + kp;
                __builtin_prefetch(pf, 0, 1);
            }
        }

        const float* A = lsA[cur];
        const float* B = lsB[cur];
#pragma unroll
        for (int k = 0; k < KC; k += 4) {
            v2f af[4], bf[2];
#pragma unroll
            for (int m = 0; m < 4; ++m)
                af[m] = *(const v2f*)&A[(wm * 64 + m * 16 + r) * LDA + k + 2 * half];
#pragma unroll
            for (int n = 0; n < 2; ++n)
                bf[n] = *(const v2f*)&B[(wn * 32 + n * 16 + r) * LDA + k + 2 * half];
#pragma unroll
            for (int m = 0; m < 4; ++m)
#pragma unroll
                for (int n = 0; n < 2; ++n)
                    acc[m][n] = __builtin_amdgcn_wmma_f32_16x16x4_f32(
                        false, af[m], false, bf[n], (short)0, acc[m][n], false, false);
        }
        // no trailing barrier needed: next iteration's wait+barrier orders everything
    }

    // epilogue: C/D layout -> G[row, col]
    float* outT = out + (size_t)blockIdx.y * C_DIM * C_DIM;
    const int rowBase = ti * BM + wm * 64;
    const int colBase = tj * BM + wn * 32;
#pragma unroll
    for (int m = 0; m < 4; ++m)
#pragma unroll
        for (int n = 0; n < 2; ++n)
#pragma unroll
            for (int v = 0; v < 8; ++v) {
                const int gr = rowBase + m * 16 + v + 8 * half;
                const int gc = colBase + n * 16 + r;
                outT[(size_t)gr * C_DIM + gc] = acc[m][n][v];
            }
}

// ---------- split-K reduction ----------
__global__ void reduce_split_kernel(const float* __restrict__ ws, float* __restrict__ out,
                                    int split, int n) {
    const int i = blockIdx.x * blockDim.x + threadIdx.x;
    if (i >= n) return;
    float s = 0.f;
    for (int p = 0; p < split; ++p) s += ws[(size_t)p * n + i];
    out[i] = s;
}

// ---------- launch ----------
extern "C" void kernel_launch(void* const* d_in, const int* in_sizes, int n_in,
                              void* d_out, int out_size, void* d_ws, size_t ws_size,
                              hipStream_t stream) {
    const float* X  = (const float*)d_in[0];
    float* out      = (float*)d_out;
    const int  C    = C_DIM;
    const int  npix = in_sizes[0] / C;            // 256*256 = 65536
    const size_t per = (size_t)C * C * sizeof(float);

    int split = 1;
    if (ws_size >= 16 * per)      split = 16;
    else if (ws_size >= 8 * per)  split = 8;
    else if (ws_size >= 4 * per)  split = 4;
    else if (ws_size >= 2 * per)  split = 2;

    const int kPerSplit = npix / split;           // multiple of KC for these splits
    float* tgt = (split > 1) ? (float*)d_ws : out;

    dim3 grid(16, split);
    gram_wmma_kernel<<<grid, NTHREADS, 0, stream>>>(X, tgt, npix, kPerSplit);

    if (split > 1) {
        const int n = C * C;
        reduce_split_kernel<<<(n + 255) / 256, 256, 0, stream>>>((const float*)d_ws, out, split, n);
    }
}